// AllReduce_56994216018563
// MI455X (gfx1250) — compile-verified
//
#include <hip/hip_runtime.h>
#include <stdint.h>

// Fused TP-allreduce (+bias +residual) + RMSNorm for MI455X (gfx1250).
// Bandwidth-bound: ~1.44 GiB moved per call -> ~62us floor @ 23.3 TB/s.
// CDNA5 path: async global->LDS b128 copies (ASYNCcnt-tracked, double
// buffered) stream the 1 GiB `x` tensor; all other traffic is 128-bit,
// with non-temporal hints on stream-once data so the 192MB L2 keeps the
// broadcast bias/norm_weight lines hot.

#define TP      8
#define TOKENS  8192
#define HIDDEN  4096
#define CHUNK   1024                        // floats of hidden dim per pipeline stage
#define NCHUNK  (HIDDEN / CHUNK)            // 4
#define THREADS 256                         // 8 wave32 waves
#define UNITS_PER_SHARD   (CHUNK * 4 / 16)               // 256 (16B units)
#define UNITS_PER_THREAD  (TP * UNITS_PER_SHARD / THREADS) // 8 async b128 per thread
#define EPS 1e-6f

typedef float v4f __attribute__((ext_vector_type(4)));
typedef int   v4i __attribute__((vector_size(16)));      // int4, matches builtin param

typedef __attribute__((address_space(1))) v4i* g4i_p;    // global int4*
typedef __attribute__((address_space(3))) v4i* l4i_p;    // LDS int4*

// generic -> AS1 is the identity VA; generic -> AS3 keeps addr[31:0] (ISA 10.2 aperture rule)
#define AS1_B128(p) ((g4i_p)(uintptr_t)(p))
#define AS3_B128(p) ((l4i_p)(uint32_t)(uintptr_t)(p))

#if __has_builtin(__builtin_amdgcn_global_load_async_to_lds_b128)
#define ASYNC_LD_B128(gp, lp) \
  __builtin_amdgcn_global_load_async_to_lds_b128(AS1_B128(gp), AS3_B128(lp), 0, 0)
#else
// CDNA5 ISA op 98: GLOBAL_LOAD_ASYNC_TO_LDS_B128, GV mode
#define ASYNC_LD_B128(gp, lp)                                              \
  asm volatile("global_load_async_to_lds_b128 %0, %1, off"                 \
               :: "v"((uint32_t)(uintptr_t)(lp)),                          \
                  "v"((uint64_t)(uintptr_t)(gp))                           \
               : "memory")
#endif

#if __has_builtin(__builtin_amdgcn_s_wait_asynccnt)
#define WAIT_ASYNC(n) __builtin_amdgcn_s_wait_asynccnt(n)
#else
#define WAIT_ASYNC(n) asm volatile("s_wait_asynccnt %0" :: "i"(n) : "memory")
#endif

__global__ __launch_bounds__(THREADS)
void fused_allreduce_rmsnorm_kernel(const float* __restrict__ x,
                                    const float* __restrict__ bias,
                                    const float* __restrict__ residual,
                                    const float* __restrict__ norm_weight,
                                    float* __restrict__ out) {
  __shared__ float lds_x[2][TP][CHUNK];     // 64 KiB double buffer (exactly)

  const int tid   = threadIdx.x;
  const int token = blockIdx.x;

  // Issue one pipeline stage: copy all TP shards of chunk `c` of this token's
  // row into LDS buffer `b`. With 256 threads and 256 16B-units per shard,
  // iteration i loads shard i, unit tid: each wave instruction moves 512
  // contiguous bytes of one shard. 8 async ops in flight per wave per stage.
  auto issue = [&](int c, int b) {
#pragma unroll
    for (int s = 0; s < TP; ++s) {          // shard == unroll index
      const float* g = x + ((size_t)s * TOKENS + token) * HIDDEN
                         + (size_t)c * CHUNK + (size_t)tid * 4;
      float* l = &lds_x[b][s][tid * 4];
      ASYNC_LD_B128(g, l);
    }
  };

  issue(0, 0);                              // prime the pipeline

  v4f   racc[NCHUNK];                       // residual_out values, kept in VGPRs
  float ss = 0.0f;                          // per-thread sum of squares

  float* const residual_out_base = out + (size_t)TOKENS * HIDDEN;

#pragma unroll
  for (int c = 0; c < NCHUNK; ++c) {
    if (c + 1 < NCHUNK) {
      issue(c + 1, (c + 1) & 1);            // overlap next stage's DMA
      WAIT_ASYNC(UNITS_PER_THREAD);         // stage c done (8 newer in flight)
    } else {
      WAIT_ASYNC(0);
    }
    __syncthreads();                        // stage c visible to all waves

    const int h0 = c * CHUNK + tid * 4;

    // 8-way shard reduction out of LDS (ds_load_b128, conflict-free layout)
    v4f sum = {0.0f, 0.0f, 0.0f, 0.0f};
#pragma unroll
    for (int s = 0; s < TP; ++s) {
      v4f v = *(const v4f*)&lds_x[c & 1][s][tid * 4];
      sum += v;
    }

    v4f bv = *(const v4f*)(bias + h0);      // broadcast: let L2/WGP$ cache it
    v4f rv = __builtin_nontemporal_load(
        (const v4f*)(residual + (size_t)token * HIDDEN + h0));

    v4f r = sum + bv + rv;                  // residual_out
    racc[c] = r;
    ss += r.x * r.x + r.y * r.y + r.z * r.z + r.w * r.w;

    __builtin_nontemporal_store(
        r, (v4f*)(residual_out_base + (size_t)token * HIDDEN + h0));

    __syncthreads();                        // before DMA reuses this buffer
  }

  // Block-wide reduction of sum-of-squares (wave32 shuffles + LDS scratch).
  // Pipeline buffer is dead past the final barrier above -> alias scratch
  // into it to stay at exactly 64 KiB of LDS.
  float* red = &lds_x[0][0][0];
#pragma unroll
  for (int o = 16; o > 0; o >>= 1) ss += __shfl_xor(ss, o, 32);
  if ((tid & 31) == 0) red[tid >> 5] = ss;
  __syncthreads();
  const int lane = tid & 31;
  float tot = (lane < (THREADS / 32)) ? red[lane] : 0.0f;
#pragma unroll
  for (int o = 16; o > 0; o >>= 1) tot += __shfl_xor(tot, o, 32);

  const float inv = rsqrtf(tot * (1.0f / (float)HIDDEN) + EPS);

  // norm_out = residual_out * inv_rms * weight  (from registers, no re-read)
#pragma unroll
  for (int c = 0; c < NCHUNK; ++c) {
    const int h0 = c * CHUNK + tid * 4;
    v4f w = *(const v4f*)(norm_weight + h0);
    v4f r = racc[c];
    v4f o4;
    o4.x = r.x * inv * w.x;
    o4.y = r.y * inv * w.y;
    o4.z = r.z * inv * w.z;
    o4.w = r.w * inv * w.w;
    __builtin_nontemporal_store(
        o4, (v4f*)(out + (size_t)token * HIDDEN + h0));
  }
}

extern "C" void kernel_launch(void* const* d_in, const int* in_sizes, int n_in,
                              void* d_out, int out_size, void* d_ws, size_t ws_size,
                              hipStream_t stream) {
  (void)in_sizes; (void)n_in; (void)out_size; (void)d_ws; (void)ws_size;
  const float* x           = (const float*)d_in[0];  // [8, 8192, 4096]
  const float* bias        = (const float*)d_in[1];  // [4096]
  const float* residual    = (const float*)d_in[2];  // [8192, 4096]
  const float* norm_weight = (const float*)d_in[3];  // [4096]
  float* out = (float*)d_out;                        // norm_out ++ residual_out

  hipLaunchKernelGGL(fused_allreduce_rmsnorm_kernel,
                     dim3(TOKENS), dim3(THREADS), 0, stream,
                     x, bias, residual, norm_weight, out);
}